// SimilarityComputation_21199958573323
// MI455X (gfx1250) — compile-verified
//
#include <hip/hip_runtime.h>
#include <hip/hip_bf16.h>

typedef __attribute__((ext_vector_type(2))) float v2f;
typedef __attribute__((ext_vector_type(4))) float v4f;
typedef __attribute__((ext_vector_type(8))) float v8f;

#define BATCH 4
#define NROW 2048
#define MROW 2048
#define CDIM 128
#define TILE 128

__global__ __launch_bounds__(256, 2)
void sim_wmma_f32_kernel(const float* __restrict__ S,
                         const float* __restrict__ T,
                         float* __restrict__ O) {
    __shared__ float s_sq[TILE];
    __shared__ float s_rn[TILE];
    __shared__ float t_sq[TILE];
    __shared__ float t_rn[TILE];

    const int b  = blockIdx.z;
    const int nb = blockIdx.y * TILE;
    const int mb = blockIdx.x * TILE;

    const float* Sb = S + (size_t)b * NROW * CDIM;
    const float* Tb = T + (size_t)b * MROW * CDIM;

    // ---- Phase 1: per-row squared norms for this tile's 128 src + 128 tgt rows ----
    {
        const int tid = threadIdx.x;
        const float* base = (tid < TILE)
            ? (Sb + (size_t)(nb + tid) * CDIM)
            : (Tb + (size_t)(mb + (tid - TILE)) * CDIM);
        float acc = 0.0f;
#pragma unroll
        for (int k = 0; k < CDIM; k += 4) {
            v4f v = *(const v4f*)(base + k);
            acc += v.x * v.x + v.y * v.y + v.z * v.z + v.w * v.w;
        }
        const float nrm = __builtin_amdgcn_sqrtf(acc);
        const float rn  = __builtin_amdgcn_rcpf(fmaxf(nrm, 1e-12f));
        if (tid < TILE) { s_sq[tid] = acc; s_rn[tid] = rn; }
        else           { t_sq[tid - TILE] = acc; t_rn[tid - TILE] = rn; }
    }
    __syncthreads();

    // ---- Phase 2: GEMM dots = S * T^T on f32 WMMA (16x16x4) ----
    const int wid   = threadIdx.x >> 5;   // 0..7
    const int lane  = threadIdx.x & 31;
    const int lhalf = lane >> 4;          // 0 or 1
    const int lmod  = lane & 15;

    // wave sub-tile: 32 rows x 64 cols; waves arranged 4 (rows) x 2 (cols)
    const int wrow = (wid & 3) * 32;      // 0,32,64,96
    const int wcol = (wid >> 2) * 64;     // 0,64

    v8f acc[2][4];
#pragma unroll
    for (int i = 0; i < 2; ++i)
#pragma unroll
        for (int j = 0; j < 4; ++j)
            acc[i][j] = (v8f){0.f, 0.f, 0.f, 0.f, 0.f, 0.f, 0.f, 0.f};

    // A fragment (16x4 f32): lane l holds row (l%16), cols k0 + 2*(l/16) + {0,1}
    const float* Aptr0 = Sb + (size_t)(nb + wrow +      lmod) * CDIM + 2 * lhalf;
    const float* Aptr1 = Sb + (size_t)(nb + wrow + 16 + lmod) * CDIM + 2 * lhalf;
    // B fragment (4x16 = T^T tile): same per-lane pattern over T's rows (cols of B)
    const float* Bptr0 = Tb + (size_t)(mb + wcol +      lmod) * CDIM + 2 * lhalf;
    const float* Bptr1 = Tb + (size_t)(mb + wcol + 16 + lmod) * CDIM + 2 * lhalf;
    const float* Bptr2 = Tb + (size_t)(mb + wcol + 32 + lmod) * CDIM + 2 * lhalf;
    const float* Bptr3 = Tb + (size_t)(mb + wcol + 48 + lmod) * CDIM + 2 * lhalf;

    // Fully unrolled K loop: all addresses become immediate offsets (<= 508B),
    // no loop-carried pointer arithmetic or branch overhead.
#pragma unroll
    for (int k0 = 0; k0 < CDIM; k0 += 4) {
        v2f a0 = *(const v2f*)(Aptr0 + k0);
        v2f a1 = *(const v2f*)(Aptr1 + k0);
        v2f b0 = *(const v2f*)(Bptr0 + k0);
        v2f b1 = *(const v2f*)(Bptr1 + k0);
        v2f b2 = *(const v2f*)(Bptr2 + k0);
        v2f b3 = *(const v2f*)(Bptr3 + k0);

        acc[0][0] = __builtin_amdgcn_wmma_f32_16x16x4_f32(false, a0, false, b0, (short)0, acc[0][0], false, false);
        acc[0][1] = __builtin_amdgcn_wmma_f32_16x16x4_f32(false, a0, false, b1, (short)0, acc[0][1], false, false);
        acc[0][2] = __builtin_amdgcn_wmma_f32_16x16x4_f32(false, a0, false, b2, (short)0, acc[0][2], false, false);
        acc[0][3] = __builtin_amdgcn_wmma_f32_16x16x4_f32(false, a0, false, b3, (short)0, acc[0][3], false, false);
        acc[1][0] = __builtin_amdgcn_wmma_f32_16x16x4_f32(false, a1, false, b0, (short)0, acc[1][0], false, false);
        acc[1][1] = __builtin_amdgcn_wmma_f32_16x16x4_f32(false, a1, false, b1, (short)0, acc[1][1], false, false);
        acc[1][2] = __builtin_amdgcn_wmma_f32_16x16x4_f32(false, a1, false, b2, (short)0, acc[1][2], false, false);
        acc[1][3] = __builtin_amdgcn_wmma_f32_16x16x4_f32(false, a1, false, b3, (short)0, acc[1][3], false, false);
    }

    // ---- Epilogue: cos = dot*rs*rt ; f = 1/(1+sqrt(max(ssq+tsq-2dot,0))) ----
#pragma unroll
    for (int i = 0; i < 2; ++i) {
        const int nloc0 = wrow + i * 16 + 8 * lhalf;   // rows for r=0..7 are consecutive
        // hoist per-row norm data out of the (j, r) loops
        float ssq_r[8], srn_r[8];
#pragma unroll
        for (int r = 0; r < 8; ++r) {
            ssq_r[r] = s_sq[nloc0 + r];
            srn_r[r] = s_rn[nloc0 + r];
        }
        const size_t row0 = ((size_t)b * NROW + (nb + nloc0)) * MROW;
#pragma unroll
        for (int j = 0; j < 4; ++j) {
            const int mloc = wcol + j * 16 + lmod;
            const float tsq = t_sq[mloc];
            const float trn = t_rn[mloc];
            float* outp = O + (row0 + (mb + mloc)) * 2;
#pragma unroll
            for (int r = 0; r < 8; ++r) {
                const float dot  = acc[i][j][r];
                const float cosv = dot * srn_r[r] * trn;
                const float dsq  = fmaxf(fmaf(-2.0f, dot, ssq_r[r] + tsq), 0.0f);
                const float dist = __builtin_amdgcn_sqrtf(dsq);
                const float fd   = __builtin_amdgcn_rcpf(1.0f + dist);
                v2f o;
                o.x = cosv;
                o.y = fd;
                __builtin_nontemporal_store(o, (v2f*)(outp + (size_t)r * (MROW * 2)));
            }
        }
    }
}

extern "C" void kernel_launch(void* const* d_in, const int* in_sizes, int n_in,
                              void* d_out, int out_size, void* d_ws, size_t ws_size,
                              hipStream_t stream) {
    (void)in_sizes; (void)n_in; (void)out_size; (void)d_ws; (void)ws_size;
    const float* src = (const float*)d_in[0];
    const float* tgt = (const float*)d_in[1];
    float* out = (float*)d_out;
    dim3 grid(MROW / TILE, NROW / TILE, BATCH);
    dim3 block(256);
    hipLaunchKernelGGL(sim_wmma_f32_kernel, grid, block, 0, stream, src, tgt, out);
}